// AdditiveAttention_69200513073705
// MI455X (gfx1250) — compile-verified
//
#include <hip/hip_runtime.h>
#include <cstdint>
#include <cstddef>

typedef __attribute__((ext_vector_type(2))) float v2f;
typedef __attribute__((ext_vector_type(8))) float v8f;

#define B_N 4
#define Q_N 1024
#define K_N 1024
#define D_N 256   // QSIZE == KSIZE
#define H_N 32
#define V_N 256

#if __has_builtin(__builtin_amdgcn_tanhf)
#define TANH(x) __builtin_amdgcn_tanhf(x)
#else
#define TANH(x) tanhf(x)
#endif

// ---------------------------------------------------------------------------
// CDNA5 async global->LDS copy (ASYNCcnt-tracked).  LDS address is the byte
// offset of our single static __shared__ block (static alloc starts at 0).
// ---------------------------------------------------------------------------
__device__ __forceinline__ void async_b128(uint32_t lds_byte, const void* g) {
  asm volatile("global_load_async_to_lds_b128 %0, %1, off"
               :: "v"(lds_byte), "v"((unsigned long long)(uintptr_t)g)
               : "memory");
}

__device__ __forceinline__ void wait_async0() {
#if __has_builtin(__builtin_amdgcn_s_wait_asynccnt)
  __builtin_amdgcn_s_wait_asynccnt(0);
#else
  asm volatile("s_wait_asynccnt 0x0" ::: "memory");
#endif
}

// ---------------------------------------------------------------------------
// Projection: out[r, 0..31] = X[r, 0..255] @ W[256,32] + bias, via f32 WMMA.
// One wave per 16-row tile, contraction in K=4 steps, 2 N-tiles (H=32).
// A layout (16x4 f32): lane L -> row L%16; VGPR0/1 hold K = 2*(L/16)+{0,1}.
// ---------------------------------------------------------------------------
__global__ __launch_bounds__(128) void proj_kernel(
    const float* __restrict__ X, const float* __restrict__ W,
    const float* __restrict__ bias, float* __restrict__ out) {
  const int lane = threadIdx.x & 31;
  const int wave = threadIdx.x >> 5;
  const int m    = lane & 15;
  const int hi   = lane >> 4;
  const int r0   = (blockIdx.x * 4 + wave) * 16;
  const float* xrow = X + (size_t)(r0 + m) * D_N;

  v8f z = {};
  v8f acc0 = z, acc1 = z;
#pragma unroll 4
  for (int k0 = 0; k0 < D_N; k0 += 4) {
    const int ka = k0 + 2 * hi;
    v2f a;  a.x  = xrow[ka];              a.y  = xrow[ka + 1];
    v2f b0; b0.x = W[ka*H_N + m];         b0.y = W[(ka+1)*H_N + m];
    v2f b1; b1.x = W[ka*H_N + 16 + m];    b1.y = W[(ka+1)*H_N + 16 + m];
    acc0 = __builtin_amdgcn_wmma_f32_16x16x4_f32(false, a, false, b0,
                                                 (short)0, acc0, false, false);
    acc1 = __builtin_amdgcn_wmma_f32_16x16x4_f32(false, a, false, b1,
                                                 (short)0, acc1, false, false);
  }
  const float bc0 = bias[m], bc1 = bias[16 + m];
#pragma unroll
  for (int r = 0; r < 8; ++r) {          // D: reg r -> row r + 8*hi, col = m
    const int row = r0 + r + 8 * hi;
    out[(size_t)row * H_N + m]      = acc0[r] + bc0;
    out[(size_t)row * H_N + 16 + m] = acc1[r] + bc1;
  }
}

// ---------------------------------------------------------------------------
// Fused additive attention.  Block = 4 waves = 4 q-tiles of one batch.
// Per 16-key chunk: fk (16x32) + V (16x256) staged to LDS asynchronously,
// scores computed with v_tanh_f32, weights = exp(score) (bounded, no max
// subtraction needed), P@V accumulated with f32 WMMA, normalize at the end.
// ---------------------------------------------------------------------------
#define FK_F 512                 // floats per fk buffer (16*32)
#define V_F  4096                // floats per V buffer (16*256)
#define SH_FK(buf) ((buf) * FK_F)
#define SH_V(buf)  (2 * FK_F + (buf) * V_F)

__global__ __launch_bounds__(128) void attn_kernel(
    const float* __restrict__ fqg, const float* __restrict__ fkg,
    const float* __restrict__ vg,  const int* __restrict__ valid,
    const float* __restrict__ wvp, const float* __restrict__ bvp,
    float* __restrict__ out) {
  __shared__ float sh[2 * FK_F + 2 * V_F];   // 36 KB

  const int tid  = threadIdx.x;
  const int lane = tid & 31;
  const int wave = tid >> 5;
  const int m    = lane & 15;
  const int hi   = lane >> 4;
  const int b    = blockIdx.x >> 4;          // 16 blocks per batch
  const int qg   = blockIdx.x & 15;
  const int row0 = (qg * 4 + wave) * 16;

  float wvr[32], fqr[32];
#pragma unroll
  for (int h = 0; h < 32; ++h) wvr[h] = wvp[h];        // uniform -> SGPRs
  const float* fqrow = fqg + ((size_t)b * Q_N + row0 + m) * H_N;
#pragma unroll
  for (int h = 0; h < 32; ++h) fqr[h] = fqrow[h];
  const float bv   = bvp[0];
  const int   vlen = valid[b];

  v8f z = {};
  v8f acc[16];                                          // 16 N-tiles of VDIM
#pragma unroll
  for (int nt = 0; nt < 16; ++nt) acc[nt] = z;
  float rsum = 0.0f;

  const float* fkbase = fkg + (size_t)b * K_N * H_N;
  const float* vbase  = vg  + (size_t)b * K_N * V_N;

  // Prefetch chunk 0: 128 threads cover 512 + 4096 floats as b128s.
  async_b128((uint32_t)((SH_FK(0) + tid * 4) * 4), fkbase + tid * 4);
#pragma unroll
  for (int i = 0; i < 8; ++i) {
    const int idx = i * 128 + tid;
    async_b128((uint32_t)((SH_V(0) + idx * 4) * 4), vbase + idx * 4);
  }

  const int NC = K_N / 16;
  for (int kc = 0; kc < NC; ++kc) {
    const int cur = kc & 1;
    wait_async0();            // this wave's staging of chunk kc is complete
    __syncthreads();          // everyone's staging visible; prev buf reusable

    if (kc + 1 < NC) {        // prefetch next chunk into the other buffer
      const int nxt = cur ^ 1;
      const float* fks = fkbase + (size_t)(kc + 1) * 16 * H_N;
      const float* vs  = vbase  + (size_t)(kc + 1) * 16 * V_N;
      async_b128((uint32_t)((SH_FK(nxt) + tid * 4) * 4), fks + tid * 4);
#pragma unroll
      for (int i = 0; i < 8; ++i) {
        const int idx = i * 128 + tid;
        async_b128((uint32_t)((SH_V(nxt) + idx * 4) * 4), vs + idx * 4);
      }
    }

    const float* fkb = sh + SH_FK(cur);
    const float* vb  = sh + SH_V(cur);

    // Scores/weights in the WMMA A-fragment layout: lane holds q-row m,
    // k-slots 4j + 2*hi + {0,1} for sub-chunk j.
    float wf[8];
#pragma unroll
    for (int j = 0; j < 4; ++j) {
#pragma unroll
      for (int t = 0; t < 2; ++t) {
        const int kk = 4 * j + 2 * hi + t;
        float s = bv;
        const float4* fr = reinterpret_cast<const float4*>(fkb + kk * H_N);
#pragma unroll
        for (int h4 = 0; h4 < 8; ++h4) {
          const float4 f = fr[h4];
          s += wvr[4*h4+0] * TANH(fqr[4*h4+0] + f.x);
          s += wvr[4*h4+1] * TANH(fqr[4*h4+1] + f.y);
          s += wvr[4*h4+2] * TANH(fqr[4*h4+2] + f.z);
          s += wvr[4*h4+3] * TANH(fqr[4*h4+3] + f.w);
        }
        const int kglob = kc * 16 + kk;
        wf[2*j + t] = (kglob < vlen) ? __expf(s) : 0.0f;  // |s| <= ~5: safe
      }
    }
    rsum += (wf[0]+wf[1]) + (wf[2]+wf[3]) + (wf[4]+wf[5]) + (wf[6]+wf[7]);

    // P(16x16) @ V(16x256) via 4 K-steps x 16 N-tiles of f32 WMMA.
#pragma unroll
    for (int j = 0; j < 4; ++j) {
      v2f a; a.x = wf[2*j]; a.y = wf[2*j + 1];
      const int kr = 4 * j + 2 * hi;
#pragma unroll
      for (int nt = 0; nt < 16; ++nt) {
        v2f bf;
        bf.x = vb[kr * V_N + nt * 16 + m];
        bf.y = vb[(kr + 1) * V_N + nt * 16 + m];
        acc[nt] = __builtin_amdgcn_wmma_f32_16x16x4_f32(
            false, a, false, bf, (short)0, acc[nt], false, false);
      }
    }
  }

  // Row sums live split across lanes m and m+16; combine, then normalize.
  const float rtot = rsum + __shfl_xor(rsum, 16, 32);
#pragma unroll
  for (int r = 0; r < 8; ++r) {          // acc reg r -> q-row r + 8*hi
    const float inv = 1.0f / __shfl(rtot, r + 8 * hi, 32);
    float* orow = out + ((size_t)b * Q_N + row0 + r + 8 * hi) * V_N;
#pragma unroll
    for (int nt = 0; nt < 16; ++nt)
      orow[nt * 16 + m] = acc[nt][r] * inv;
  }
}

// ---------------------------------------------------------------------------
extern "C" void kernel_launch(void* const* d_in, const int* in_sizes, int n_in,
                              void* d_out, int out_size, void* d_ws,
                              size_t ws_size, hipStream_t stream) {
  const float* keys    = (const float*)d_in[0];
  const float* queries = (const float*)d_in[1];
  const float* values  = (const float*)d_in[2];
  const int*   vlens   = (const int*)  d_in[3];
  const float* W_q     = (const float*)d_in[4];
  const float* b_q     = (const float*)d_in[5];
  const float* W_k     = (const float*)d_in[6];
  const float* b_k     = (const float*)d_in[7];
  const float* w_v     = (const float*)d_in[8];
  const float* b_v     = (const float*)d_in[9];
  float* out = (float*)d_out;

  float* fq = (float*)d_ws;              // B*Q*H = 128K floats
  float* fk = fq + (size_t)B_N * Q_N * H_N;

  // 4096 rows each -> 256 row-tiles -> 64 blocks of 4 waves.
  proj_kernel<<<64, 128, 0, stream>>>(queries, W_q, b_q, fq);
  proj_kernel<<<64, 128, 0, stream>>>(keys,    W_k, b_k, fk);
  // 4 batches * 16 blocks (4 q-tiles each).
  attn_kernel<<<64, 128, 0, stream>>>(fq, fk, values, vlens, w_v, b_v, out);
}